// MyGRUModel_7378753814825
// MI455X (gfx1250) — compile-verified
//
#include <hip/hip_runtime.h>
#include <hip/hip_bf16.h>
#include <stdint.h>

typedef float v2f __attribute__((ext_vector_type(2)));
typedef float v8f __attribute__((ext_vector_type(8)));

#define NFEAT 10
#define HID   30
#define TT    512
#define SEQ   1536   // 3*T
#define BB    2048
#define NTILE (BB/16)   // 128 batch tiles of 16

__device__ __forceinline__ float fast_exp(float x) {
    return __builtin_amdgcn_exp2f(x * 1.4426950408889634f);
}
__device__ __forceinline__ float fast_sigmoid(float x) {
    return __builtin_amdgcn_rcpf(1.0f + fast_exp(-x));
}
__device__ __forceinline__ float fast_tanh(float x) {
    return 2.0f * __builtin_amdgcn_rcpf(1.0f + fast_exp(-2.0f * x)) - 1.0f;
}

// ---------------------------------------------------------------------------
// Kernel 1: build permuted features x[b,s,d] = feat_flat[b, d*1536+s] directly
// in the WMMA f32 16x16x4 A-operand swizzle:
//   xswz[(((bt*SEQ)+s)*3 + kc)*32 + lane] = float2 {A(K=k0), A(K=k0+1)}
//   with M = lane%16 (batch row in tile), k0 = 4*kc + 2*(lane/16), K pad 10->12
// ---------------------------------------------------------------------------
__global__ void feat_permute_kernel(const int* __restrict__ ai,
                                    const int* __restrict__ aj,
                                    const float* __restrict__ dist,
                                    const float* __restrict__ emb,
                                    float2* __restrict__ xswz) {
    int gid  = blockIdx.x * blockDim.x + threadIdx.x;
    int lane = gid & 31;
    int kc   = (gid >> 5) % 3;
    int s    = (gid / 96) % SEQ;
    int bt   = gid / (96 * SEQ);
    int b    = bt * 16 + (lane & 15);
    int d0   = 4 * kc + 2 * (lane >> 4);

    float v[2];
#pragma unroll
    for (int j = 0; j < 2; ++j) {
        int d = d0 + j;
        float r = 0.0f;
        if (d < NFEAT) {
            int idx = d * SEQ + s;           // position in flat [T*30] feature row
            int t   = idx / HID;             // time step
            int c   = idx - t * HID;         // channel within feat[.,.,30]
            int a_i = ai[b * TT + t];
            if (c < 20) {
                int a  = (c < 10) ? a_i : aj[b * TT + t];
                int cc = (c < 10) ? c : (c - 10);
                r = (a != 0) ? emb[a * NFEAT + cc] : 0.0f;
            } else if (a_i != 0) {
                float ctr = 0.7f * (float)(c - 19);   // centers 0.7*(1..10)
                float df  = ctr - dist[b * TT + t];
                r = fast_exp(-df * df);               // GAMMA = 1
            }
        }
        v[j] = r;
    }
    xswz[gid] = make_float2(v[0], v[1]);
}

// ---------------------------------------------------------------------------
// Kernel 2: GRU scan. One wave32 per 16-batch tile.
// Gate dim padded 90 -> 96 (r:0-31, z:32-63, n:64-95), hidden padded 30 -> 32.
// r,z tiles: gi and gh fused into ONE accumulator chain (C starts at inline 0,
// biases folded into the gate phase). n tiles keep separate gi/gh accumulators
// because of the r*h_n term. 66 v_wmma_f32_16x16x4_f32 per step.
// h transposed D-layout -> A-layout through wave-private LDS each step.
// ---------------------------------------------------------------------------
__global__ void __launch_bounds__(32)
gru_scan_kernel(const float* __restrict__ W_ih, const float* __restrict__ W_hh,
                const float* __restrict__ b_ih, const float* __restrict__ b_hh,
                const float* __restrict__ W_out, const float* __restrict__ b_out,
                const float2* __restrict__ xswz, float* __restrict__ out) {
    __shared__ float hs[16 * 34];   // [batch 16][hidden 32 + pad 2]

    const int lane = threadIdx.x & 31;
    const int nlo  = lane & 15;
    const int half = lane >> 4;
    const int bt   = blockIdx.x;

    // Per-lane biases, folded into the gate phase (not into WMMA C):
    //   bsum[t], t=0..3 : b_ih + b_hh over padded r,z columns
    //   bihn/bhhn[t]    : separate n-gate biases (needed due to r*h_n)
    float bsum[4], bihn[2], bhhn[2];
#pragma unroll
    for (int t = 0; t < 4; ++t) {
        int p = 16 * t + nlo, g = p >> 5, q = p & 31;
        bsum[t] = (q < HID) ? (b_ih[g * HID + q] + b_hh[g * HID + q]) : 0.0f;
    }
#pragma unroll
    for (int t = 0; t < 2; ++t) {
        int q = 16 * t + nlo;
        bihn[t] = (q < HID) ? b_ih[2 * HID + q] : 0.0f;
        bhhn[t] = (q < HID) ? b_hh[2 * HID + q] : 0.0f;
    }

    // Resident B-operand weight tiles. B element (K,N) lives at
    // vgpr = K%2 (the float2 slot), lane = N + 16*((K%4)/2), chunk = K/4.
    v2f wih[3][6];
#pragma unroll
    for (int kc = 0; kc < 3; ++kc)
#pragma unroll
        for (int t = 0; t < 6; ++t) {
            int p = 16 * t + nlo, g = p >> 5, q = p & 31;
#pragma unroll
            for (int j = 0; j < 2; ++j) {
                int K = 4 * kc + 2 * half + j;
                wih[kc][t][j] = (q < HID && K < NFEAT) ? W_ih[(g * HID + q) * NFEAT + K] : 0.0f;
            }
        }
    v2f whh[8][6];
#pragma unroll
    for (int kc = 0; kc < 8; ++kc)
#pragma unroll
        for (int t = 0; t < 6; ++t) {
            int p = 16 * t + nlo, g = p >> 5, q = p & 31;
#pragma unroll
            for (int j = 0; j < 2; ++j) {
                int K = 4 * kc + 2 * half + j;
                whh[kc][t][j] = (q < HID && K < HID) ? W_hh[(g * HID + q) * HID + K] : 0.0f;
            }
        }

    // h state, D-layout: hD[t][v] holds h[b = v + 8*half, m = 16*t + nlo]
    v8f hD[2];
#pragma unroll
    for (int t = 0; t < 2; ++t)
#pragma unroll
        for (int v = 0; v < 8; ++v) hD[t][v] = 0.0f;

    const float2* xp = xswz + (size_t)bt * SEQ * 96 + lane;

    for (int s = 0; s < SEQ; ++s) {
        // x A-operands for this step (coalesced 256B per chunk)
        v2f xa[3];
#pragma unroll
        for (int kc = 0; kc < 3; ++kc) {
            float2 xv = xp[kc * 32];
            xa[kc][0] = xv.x; xa[kc][1] = xv.y;
        }
        xp += 96;

        // Transpose h: D-layout regs -> LDS [b][m] -> A-layout regs.
#pragma unroll
        for (int t = 0; t < 2; ++t)
#pragma unroll
            for (int v = 0; v < 8; ++v)
                hs[(v + 8 * half) * 34 + 16 * t + nlo] = hD[t][v];
        asm volatile("s_wait_dscnt 0x0" ::: "memory");   // same-wave DS RAW
        v2f ha[8];
#pragma unroll
        for (int kc = 0; kc < 8; ++kc) {
            float2 hv = *(const float2*)&hs[nlo * 34 + 4 * kc + 2 * half];
            ha[kc][0] = hv.x; ha[kc][1] = hv.y;
        }

        // Fused r,z accumulators (tiles 0..3): gi chain then gh chain, C0 = 0.
        v8f S[4];
#pragma unroll
        for (int t = 0; t < 4; ++t) {
            v8f acc = {};   // first WMMA of chain gets inline SRC2 = 0
#pragma unroll
            for (int kc = 0; kc < 3; ++kc)
                acc = __builtin_amdgcn_wmma_f32_16x16x4_f32(
                    false, xa[kc], false, wih[kc][t], (short)0, acc, false, false);
#pragma unroll
            for (int kc = 0; kc < 8; ++kc)
                acc = __builtin_amdgcn_wmma_f32_16x16x4_f32(
                    false, ha[kc], false, whh[kc][t], (short)0, acc, false, false);
            S[t] = acc;
        }
        // Separate n-gate accumulators (tiles 4,5).
        v8f Gn[2], Hn[2];
#pragma unroll
        for (int t = 0; t < 2; ++t) {
            v8f gi = {};
#pragma unroll
            for (int kc = 0; kc < 3; ++kc)
                gi = __builtin_amdgcn_wmma_f32_16x16x4_f32(
                    false, xa[kc], false, wih[kc][4 + t], (short)0, gi, false, false);
            Gn[t] = gi;
            v8f gh = {};
#pragma unroll
            for (int kc = 0; kc < 8; ++kc)
                gh = __builtin_amdgcn_wmma_f32_16x16x4_f32(
                    false, ha[kc], false, whh[kc][4 + t], (short)0, gh, false, false);
            Hn[t] = gh;
        }

        // Gate phase (biases folded in here). Identical element layouts.
#pragma unroll
        for (int t = 0; t < 2; ++t)
#pragma unroll
            for (int v = 0; v < 8; ++v) {
                float r = fast_sigmoid(S[t][v]     + bsum[t]);
                float z = fast_sigmoid(S[2 + t][v] + bsum[2 + t]);
                float n = fast_tanh((Gn[t][v] + bihn[t]) + r * (Hn[t][v] + bhhn[t]));
                hD[t][v] = (1.0f - z) * n + z * hD[t][v];
            }
    }

    // out[b] = relu(h . W_out + b_out); reduce over m (lanes within each half)
    float w0 = W_out[nlo];
    float w1 = (nlo < HID - 16) ? W_out[16 + nlo] : 0.0f;
    float bo = b_out[0];
#pragma unroll
    for (int v = 0; v < 8; ++v) {
        float part = hD[0][v] * w0 + hD[1][v] * w1;
#pragma unroll
        for (int mask = 1; mask < 16; mask <<= 1)
            part += __shfl_xor(part, mask, 32);
        if (nlo == 0) {
            float o = part + bo;
            out[bt * 16 + v + 8 * half] = o > 0.0f ? o : 0.0f;
        }
    }
}

// ---------------------------------------------------------------------------
extern "C" void kernel_launch(void* const* d_in, const int* in_sizes, int n_in,
                              void* d_out, int out_size, void* d_ws, size_t ws_size,
                              hipStream_t stream) {
    (void)in_sizes; (void)n_in; (void)out_size; (void)ws_size;
    const int*   atom_i = (const int*)  d_in[0];
    const int*   atom_j = (const int*)  d_in[1];
    const float* dist   = (const float*)d_in[2];
    const float* emb    = (const float*)d_in[3];
    const float* W_ih   = (const float*)d_in[4];
    const float* W_hh   = (const float*)d_in[5];
    const float* b_ih   = (const float*)d_in[6];
    const float* b_hh   = (const float*)d_in[7];
    const float* W_out  = (const float*)d_in[8];
    const float* b_out  = (const float*)d_in[9];
    float*  out  = (float*)d_out;
    float2* xswz = (float2*)d_ws;   // NTILE*SEQ*3*32 float2 = ~151 MB

    const int n1 = NTILE * SEQ * 96;          // float2 elements
    feat_permute_kernel<<<n1 / 256, 256, 0, stream>>>(atom_i, atom_j, dist, emb, xswz);
    gru_scan_kernel<<<NTILE, 32, 0, stream>>>(W_ih, W_hh, b_ih, b_hh, W_out, b_out,
                                              xswz, out);
}